// GQHAN_75170517615005
// MI455X (gfx1250) — compile-verified
//
#include <hip/hip_runtime.h>
#include <math.h>

typedef __attribute__((ext_vector_type(2))) float v2f;
typedef __attribute__((ext_vector_type(8))) float v8f;

// ---------------------------------------------------------------------------
// Kernel 1: build Q = M^T D M (8x8) from the circuit parameters.
// M = 16x8 real circuit matrix (circuit applied to 8 data-basis states),
// D = diag(+1 if wire3 bit == 0 else -1).
// State index: idx = anc*8 + w1*4 + w2*2 + w3  (anc bit=3, data bits 2,1,0).
// One block of 64 threads; threads 0..7 simulate basis columns, threads
// 0..63 each produce one Q entry. One-shot setup cost, negligible.
// ---------------------------------------------------------------------------
__global__ void gqhan_build_q(const float* __restrict__ theta_fo,
                              const float* __restrict__ theta_ado,
                              float* __restrict__ Qout) {
  __shared__ float Mm[16][8];
  const int t = threadIdx.x;

  if (t < 8) {
    float st[16];
#pragma unroll
    for (int i = 0; i < 16; ++i) st[i] = 0.0f;
    st[t] = 1.0f;  // ancilla=0, data basis state t

    // --- Flexible oracle: RY(th) on anc, flip amp (anc=1,data=b), RY(-th) ---
    for (int b = 0; b < 8; ++b) {
      const float th = 0.5f * theta_fo[b];
      const float c = cosf(th), s = sinf(th);
#pragma unroll
      for (int i = 0; i < 8; ++i) {  // RY(+th) on ancilla (stride 8)
        const float a0 = st[i], a1 = st[i + 8];
        st[i] = c * a0 - s * a1;
        st[i + 8] = s * a0 + c * a1;
      }
      st[8 + b] = -st[8 + b];        // bitstring-controlled phase flip
#pragma unroll
      for (int i = 0; i < 8; ++i) {  // RY(-th) on ancilla
        const float a0 = st[i], a1 = st[i + 8];
        st[i] = c * a0 + s * a1;
        st[i + 8] = -s * a0 + c * a1;
      }
    }

    auto had = [&](int wb) {
      const int str = 1 << wb;
      const float r = 0.70710678118654752440f;
#pragma unroll
      for (int i = 0; i < 16; ++i) {
        if (i & str) continue;
        const float a0 = st[i], a1 = st[i + str];
        st[i] = r * (a0 + a1);
        st[i + str] = r * (a0 - a1);
      }
    };
    auto cry = [&](int cb, int tb, float th) {
      const float c = cosf(0.5f * th), s = sinf(0.5f * th);
      const int cs = 1 << cb, ts = 1 << tb;
#pragma unroll
      for (int i = 0; i < 16; ++i) {
        if (!(i & cs) || (i & ts)) continue;
        const float a0 = st[i], a1 = st[i + ts];
        st[i] = c * a0 - s * a1;
        st[i + ts] = s * a0 + c * a1;
      }
    };

    // --- ADO with X sandwich ---
    had(2); had(1); had(0);
    const int wbit[3] = {2, 1, 0};  // data wires 1,2,3 -> idx bits 2,1,0
#pragma unroll
    for (int i = 0; i < 3; ++i) cry(wbit[i], wbit[(i + 1) % 3], theta_ado[i]);
    st[0] = -st[0];  // phase flip where all data wires == 0
    st[8] = -st[8];
#pragma unroll
    for (int i = 0; i < 3; ++i) cry(wbit[i], wbit[(i + 1) % 3], theta_ado[3 + i]);
    had(2); had(1); had(0);

#pragma unroll
    for (int k = 0; k < 16; ++k) Mm[k][t] = st[k];
  }

  __syncthreads();

  if (t < 64) {
    const int i = t >> 3, j = t & 7;
    float q = 0.0f;
#pragma unroll
    for (int k = 0; k < 16; ++k) {
      const float sgn = (k & 1) ? -1.0f : 1.0f;  // <Z> on wire 3 (idx bit 0)
      q += sgn * Mm[k][i] * Mm[k][j];
    }
    Qout[t] = q;  // row-major 8x8
  }
}

// ---------------------------------------------------------------------------
// Half-wave swap (lane ^ 16) via ds_swizzle SWAPX16: xor=0x10,or=0,and=0x1f
// -> imm = (0x10<<10) | 0x1f = 0x401F. Single DS op, no guards needed.
// ---------------------------------------------------------------------------
__device__ __forceinline__ float swap16(float v) {
  return __int_as_float(
      __builtin_amdgcn_ds_swizzle(__float_as_int(v), 0x401F));
}

// ---------------------------------------------------------------------------
// Kernel 2: out[s] = x_s^T Q x_s / (x_s^T x_s), transposed-GEMM formulation:
//   D = G * X^T,  G = [Q ; I8] (16x8, A operand), X^T = 8x16 (B operand)
// D column n = sample n: lanes 0..15 hold (Q x_n)[r] in VGPR r, lanes 16..31
// hold x_n[r]. One SWAPX16 per VGPR pairs y_r with x_r in-lane; then
//   q = sum_r y_r*x_r,  n = sum_r x_r^2,  out = q * rcp(n)
// is 16 per-lane FMAs. Lanes 0..15 store one coalesced 64B burst per tile.
// Each wave processes two 16-sample tiles (A operand reused).
// ---------------------------------------------------------------------------
__global__ __launch_bounds__(256) void gqhan_qform(const float* __restrict__ x,
                                                   const float* __restrict__ Q,
                                                   float* __restrict__ out,
                                                   int B) {
  const int lane = threadIdx.x & 31;
  const int l = lane & 15;       // A-row (M) / B-col (N) index
  const int h = lane >> 4;       // half-wave select
  const int wave = blockIdx.x * (blockDim.x >> 5) + (threadIdx.x >> 5);
  const int s0 = wave << 5;      // 32 samples per wave (2 tiles)
  if (s0 >= B) return;           // wave-uniform exit: EXEC stays all-ones

  // ---- A operand: G[m][k] = (m<8) ? Q[m][k] : (k==m-8).  Layout:
  // chunk0 VGPR0 = {M=l, K=2h}, VGPR1 = {M=l, K=2h+1}; chunk1: K += 4.
  const int ka = 2 * h, kb = 2 * h + 1, kc = 4 + 2 * h, kd = 5 + 2 * h;
  const int j = l - 8;
  v2f a0, a1;
  a0.x = (l < 8) ? Q[l * 8 + ka] : ((j == ka) ? 1.0f : 0.0f);
  a0.y = (l < 8) ? Q[l * 8 + kb] : ((j == kb) ? 1.0f : 0.0f);
  a1.x = (l < 8) ? Q[l * 8 + kc] : ((j == kc) ? 1.0f : 0.0f);
  a1.y = (l < 8) ? Q[l * 8 + kd] : ((j == kd) ? 1.0f : 0.0f);

#pragma unroll
  for (int t = 0; t < 2; ++t) {
    // ---- B operand: X^T[k][n] = X[s0+16t+n, k]; per-lane coalesced f2 loads
    int sr = s0 + 16 * t + l;
    if (sr >= B) sr = B - 1;  // clamp tail rows (stores guarded below)
    const float* row = x + (size_t)sr * 8;
    const v2f b0 = *(const v2f*)(row + 2 * h);      // rows K=2h, 2h+1
    const v2f b1 = *(const v2f*)(row + 4 + 2 * h);  // rows K=4+2h, 5+2h

    // ---- D = G * X^T over K=8 via two K=4 WMMAs ----
    v8f acc = {};
    acc = __builtin_amdgcn_wmma_f32_16x16x4_f32(false, a0, false, b0,
                                                (short)0, acc, false, false);
    acc = __builtin_amdgcn_wmma_f32_16x16x4_f32(false, a1, false, b1,
                                                (short)0, acc, false, false);

    // ---- In-lane reduction: pair (Qx)[r] with x[r] via SWAPX16 ----
    float q = 0.0f, nn = 0.0f;
#pragma unroll
    for (int r = 0; r < 8; ++r) {
      const float d = acc[r];            // lanes<16: y_r ; lanes>=16: x_r
      const float p = swap16(d);         // lanes<16: x_r ; lanes>=16: y_r
      const float xv = (lane < 16) ? p : d;
      q = fmaf(d, p, q);                 // sum_r y_r * x_r  (both halves)
      nn = fmaf(xv, xv, nn);             // sum_r x_r^2
    }
    const float res = q * __builtin_amdgcn_rcpf(nn);

    if (lane < 16) {
      const int sidx = s0 + 16 * t + lane;
      if (sidx < B) out[sidx] = res;     // one coalesced 64B store per tile
    }
  }
}

// ---------------------------------------------------------------------------
// Launch: inputs are [x_batch (B*8 f32), theta_fo (8 f32), theta_ado (6 f32)],
// output B f32. Workspace holds the 8x8 Q matrix (256 bytes).
// ---------------------------------------------------------------------------
extern "C" void kernel_launch(void* const* d_in, const int* in_sizes, int n_in,
                              void* d_out, int out_size, void* d_ws,
                              size_t ws_size, hipStream_t stream) {
  const float* x = (const float*)d_in[0];
  const float* theta_fo = (const float*)d_in[1];
  const float* theta_ado = (const float*)d_in[2];
  float* out = (float*)d_out;
  float* Q = (float*)d_ws;
  const int B = in_sizes[0] / 8;

  hipLaunchKernelGGL(gqhan_build_q, dim3(1), dim3(64), 0, stream,
                     theta_fo, theta_ado, Q);

  const int samplesPerBlock = 256;  // 256 threads = 8 waves * 32 samples
  const int grid = (B + samplesPerBlock - 1) / samplesPerBlock;
  hipLaunchKernelGGL(gqhan_qform, dim3(grid), dim3(256), 0, stream,
                     x, Q, out, B);
}